// ObjectCondensationLoss_83786222011250
// MI455X (gfx1250) — compile-verified
//
#include <hip/hip_runtime.h>
#include <math.h>

// Problem constants (match reference)
#define Bn 16
#define Nn 65536
#define Dn 24
#define Kn 512

// Workspace layout (units of 4 bytes)
enum : int {
  O_CNT = 0,                 // u32 seg counts        [B*K]
  O_CPN = O_CNT + Bn * Kn,   // u32 cp counts         [B*K]
  O_MAX = O_CPN + Bn * Kn,   // u32 monotone-key max  [B*K]
  O_FCP = O_MAX + Bn * Kn,   // u32 first cp index    [B*K]
  O_CPL = O_FCP + Bn * Kn,   // f32 cp logit sum      [B*K]
  O_DEN = O_CPL + Bn * Kn,   // f32 softmax denom     [B*K]
  O_D2  = O_DEN + Bn * Kn,   // f32 d2 sums           [B*K]
  O_M   = O_D2  + Bn * Kn,   // f32 has_cp mask       [B*K]
  O_EMB = O_M   + Bn * Kn,   // f32 cp embeddings     [B*K*D]
  O_SC  = O_EMB + Bn * Kn * Dn, // f32 per-batch scalars [B*16]
  O_END = O_SC + Bn * 16
};
// scalars per batch: 0 pos_m, 1 neg_m, 2 pos_cnt, 3 neg_cnt,
//                    4 ce_sum, 5 slice_cnt, 6 attraction, 7 mcount, 8 rep_sum

typedef float v2f __attribute__((ext_vector_type(2)));
typedef float v8f __attribute__((ext_vector_type(8)));

__device__ __forceinline__ float nan0(float x) { return isfinite(x) ? x : 0.0f; }
__device__ __forceinline__ float clip20(float x) { return fminf(fmaxf(x, -20.f), 20.f); }

// Monotone float->u32 key so atomicMax on u32 == float max; key 0 < any real key.
__device__ __forceinline__ unsigned fkey(float f) {
  unsigned u = __float_as_uint(f);
  return (u & 0x80000000u) ? ~u : (u | 0x80000000u);
}
__device__ __forceinline__ float fdec(unsigned k) {
  unsigned u = (k & 0x80000000u) ? (k & 0x7FFFFFFFu) : ~k;
  return __uint_as_float(u);
}

__device__ __forceinline__ float wred(float v) {
  for (int o = 16; o > 0; o >>= 1) v += __shfl_down(v, o);
  return v;
}

// ---------------- kernel 0: init workspace ----------------
__global__ void k_init(unsigned* ws) {
  int i = blockIdx.x * blockDim.x + threadIdx.x;
  if (i >= O_END) return;
  unsigned v = 0;
  if (i >= O_FCP && i < O_FCP + Bn * Kn) v = Nn;  // first_cp sentinel
  ws[i] = v;
}

// ---------------- kernel 1: stats pass A ----------------
// grid = B*64, block = 256; each block handles 1024 hits of one batch.
__global__ __launch_bounds__(256) void k_statsA(const float* __restrict__ beta,
                                                const int* __restrict__ sid,
                                                const int* __restrict__ cp,
                                                float* wsf, unsigned* wsu) {
  __shared__ unsigned s_cnt[Kn], s_cp[Kn], s_max[Kn], s_fcp[Kn];
  __shared__ float s_cpl[Kn];
  __shared__ float s_red[32];
  const int tid = threadIdx.x;
  for (int k = tid; k < Kn; k += 256) {
    s_cnt[k] = 0u; s_cp[k] = 0u; s_max[k] = 0u; s_fcp[k] = 0xFFFFFFFFu; s_cpl[k] = 0.f;
  }
  __syncthreads();
  const int b = blockIdx.x >> 6;
  const int hit0 = (blockIdx.x & 63) * (Nn / 64);
  float pm = 0.f, nm = 0.f, pc = 0.f, nc = 0.f;
#pragma unroll
  for (int i = 0; i < (Nn / 64) / 256; i++) {
    int hit = hit0 + i * 256 + tid;
    int g = b * Nn + hit;
    float bv = nan0(beta[g]);
    int s = sid[g];
    int c = cp[g];
    float bc = clip20(bv);
    float logit = bc / 0.7f;                       // TAU
    atomicAdd(&s_cnt[s], 1u);
    atomicMax(&s_max[s], fkey(logit));
    float bp = 1.f / (1.f + expf(-bc));            // sigmoid
    if (c == 1) {
      atomicAdd(&s_cp[s], 1u);
      atomicMin(&s_fcp[s], (unsigned)hit);
      atomicAdd(&s_cpl[s], logit);
      pm += fmaxf(0.7f - bp, 0.f);                 // THRESHOLD+MARGIN
      pc += 1.f;
    } else {
      nm += fmaxf(bp - 0.3f, 0.f);                 // THRESHOLD-MARGIN
      nc += 1.f;
    }
  }
  __syncthreads();
  for (int k = tid; k < Kn; k += 256) {
    int o = b * Kn + k;
    if (s_cnt[k]) atomicAdd(&wsu[O_CNT + o], s_cnt[k]);
    if (s_cp[k]) { atomicAdd(&wsu[O_CPN + o], s_cp[k]); atomicAdd(&wsf[O_CPL + o], s_cpl[k]); }
    if (s_max[k]) atomicMax(&wsu[O_MAX + o], s_max[k]);
    if (s_fcp[k] != 0xFFFFFFFFu) atomicMin(&wsu[O_FCP + o], s_fcp[k]);
  }
  pm = wred(pm); nm = wred(nm); pc = wred(pc); nc = wred(nc);
  int wv = tid >> 5, ln = tid & 31;
  if (ln == 0) { s_red[wv] = pm; s_red[8 + wv] = nm; s_red[16 + wv] = pc; s_red[24 + wv] = nc; }
  __syncthreads();
  if (tid == 0) {
    float a = 0, b2 = 0, c2 = 0, d2 = 0;
    for (int i = 0; i < 8; i++) { a += s_red[i]; b2 += s_red[8 + i]; c2 += s_red[16 + i]; d2 += s_red[24 + i]; }
    float* sc = wsf + O_SC + b * 16;
    atomicAdd(&sc[0], a); atomicAdd(&sc[1], b2); atomicAdd(&sc[2], c2); atomicAdd(&sc[3], d2);
  }
}

// ---------------- kernel 2: stats pass B (softmax denominator) ----------------
__global__ __launch_bounds__(256) void k_statsB(const float* __restrict__ beta,
                                                const int* __restrict__ sid,
                                                float* wsf, const unsigned* wsu) {
  __shared__ float s_mx[Kn];
  __shared__ float s_den[Kn];
  const int tid = threadIdx.x;
  const int b = blockIdx.x >> 6;
  for (int k = tid; k < Kn; k += 256) {
    unsigned mk = wsu[O_MAX + b * Kn + k];
    s_mx[k] = mk ? fdec(mk) : 0.f;
    s_den[k] = 0.f;
  }
  __syncthreads();
  const int hit0 = (blockIdx.x & 63) * (Nn / 64);
#pragma unroll
  for (int i = 0; i < (Nn / 64) / 256; i++) {
    int g = b * Nn + hit0 + i * 256 + tid;
    float logit = clip20(nan0(beta[g])) / 0.7f;
    int s = sid[g];
    atomicAdd(&s_den[s], expf(logit - s_mx[s]));
  }
  __syncthreads();
  for (int k = tid; k < Kn; k += 256)
    if (s_den[k] != 0.f) atomicAdd(&wsf[O_DEN + b * Kn + k], s_den[k]);
}

// ---------------- kernel 3: gather CP embeddings ----------------
__global__ __launch_bounds__(512) void k_gather(const float* __restrict__ embed,
                                                float* wsf, const unsigned* wsu) {
  const int b = blockIdx.x;
  const int k = threadIdx.x;
  unsigned fcp = wsu[O_FCP + b * Kn + k];
  unsigned row = fcp < (Nn - 1) ? fcp : (Nn - 1);
  const float* src = embed + ((size_t)b * Nn + row) * Dn;
  float* dst = wsf + O_EMB + ((size_t)b * Kn + k) * Dn;
#pragma unroll
  for (int d = 0; d < Dn; d++) dst[d] = nan0(src[d]);
}

// ---------------- kernel 4: attraction (streams 96 MB of embed) ----------------
__global__ __launch_bounds__(256) void k_attract(const float* __restrict__ embed,
                                                 const int* __restrict__ sid,
                                                 float* wsf) {
  __shared__ float s_e[Kn * Dn];   // 48 KB
  __shared__ float s_d2[Kn];
  const int tid = threadIdx.x;
  const int b = blockIdx.x >> 6;
  const float* ce_src = wsf + O_EMB + (size_t)b * Kn * Dn;
  for (int i = tid; i < Kn * Dn; i += 256) s_e[i] = ce_src[i];
  for (int k = tid; k < Kn; k += 256) s_d2[k] = 0.f;
  __syncthreads();
  const int hit0 = (blockIdx.x & 63) * (Nn / 64);
#pragma unroll
  for (int i = 0; i < (Nn / 64) / 256; i++) {
    int g = b * Nn + hit0 + i * 256 + tid;
    int s = sid[g];
    const float4* row = reinterpret_cast<const float4*>(embed + (size_t)g * Dn);
    const float* ce = &s_e[s * Dn];
    float d2 = 0.f;
#pragma unroll
    for (int q = 0; q < 6; q++) {
      float4 v = row[q];
      float e0 = nan0(v.x) - ce[q * 4 + 0];
      float e1 = nan0(v.y) - ce[q * 4 + 1];
      float e2 = nan0(v.z) - ce[q * 4 + 2];
      float e3 = nan0(v.w) - ce[q * 4 + 3];
      d2 += e0 * e0 + e1 * e1 + e2 * e2 + e3 * e3;
    }
    atomicAdd(&s_d2[s], fminf(d2, 50.f));
  }
  __syncthreads();
  for (int k = tid; k < Kn; k += 256)
    if (s_d2[k] != 0.f) atomicAdd(&wsf[O_D2 + b * Kn + k], s_d2[k]);
}

// ---------------- kernel 5: per-segment finalize (CE, has_cp, attraction) ----------------
__global__ __launch_bounds__(512) void k_finalize(float* wsf, const unsigned* wsu) {
  __shared__ float r[16][4];
  const int b = blockIdx.x;
  const int k = threadIdx.x;
  const int o = b * Kn + k;
  unsigned cnt = wsu[O_CNT + o];
  unsigned cpn = wsu[O_CPN + o];
  unsigned fcp = wsu[O_FCP + o];
  unsigned mk  = wsu[O_MAX + o];
  float den = wsf[O_DEN + o];
  float cpl = wsf[O_CPL + o];
  float d2s = wsf[O_D2 + o];
  bool present = cnt > 0;
  bool valid = present && (cpn == 1);
  float mx = present ? fdec(mk) : 0.f;
  float p = expf(cpl - mx) / fmaxf(den, 1e-30f);
  float li = -logf(p + 1e-9f);
  float ce = valid ? li : 0.f;
  float sc = valid ? 1.f : 0.f;
  bool has = (fcp < Nn) && present;
  float attr = has ? d2s / fmaxf((float)cnt, 1.f) : 0.f;
  float m = has ? 1.f : 0.f;
  wsf[O_M + o] = m;
  ce = wred(ce); sc = wred(sc); attr = wred(attr); m = wred(m);
  int wv = k >> 5, ln = k & 31;
  if (ln == 0) { r[wv][0] = ce; r[wv][1] = sc; r[wv][2] = attr; r[wv][3] = m; }
  __syncthreads();
  if (k == 0) {
    float a = 0, b2 = 0, c2 = 0, d2 = 0;
    for (int i = 0; i < 16; i++) { a += r[i][0]; b2 += r[i][1]; c2 += r[i][2]; d2 += r[i][3]; }
    float* sp = wsf + O_SC + b * 16;
    sp[4] = a; sp[5] = b2; sp[6] = c2; sp[7] = d2;
  }
}

// ---------------- kernel 6: repulsion via WMMA f32 16x16x4 ----------------
// Gram = Ecp * Ecp^T per batch (512x512, K-dim 24). 8 waves x 128 tiles each.
__global__ __launch_bounds__(256) void k_repulse(float* wsf) {
  __shared__ float s_e[Kn * Dn];   // 48 KB
  __shared__ float s_sq[Kn];
  __shared__ float s_m[Kn];
  __shared__ float s_red[8];
  const int tid = threadIdx.x;
  const int b = blockIdx.x;
  const float* ce_src = wsf + O_EMB + (size_t)b * Kn * Dn;
  for (int i = tid; i < Kn * Dn; i += 256) s_e[i] = ce_src[i];
  for (int k = tid; k < Kn; k += 256) s_m[k] = wsf[O_M + b * Kn + k];
  __syncthreads();
  for (int k = tid; k < Kn; k += 256) {
    float sq = 0.f;
#pragma unroll
    for (int d = 0; d < Dn; d++) { float e = s_e[k * Dn + d]; sq += e * e; }
    s_sq[k] = sq;
  }
  __syncthreads();
  const int lane = tid & 31, wv = tid >> 5;
  const int rm = lane & 15;            // M (for A) / N (for B) within tile
  const int kb = (lane >> 4) << 1;     // K sub-offset {0,2}
  float lsum = 0.f;
  for (int t = wv; t < 32 * 32; t += 8) {   // uniform per wave -> EXEC all ones
    const int i0 = (t >> 5) << 4;
    const int j0 = (t & 31) << 4;
    v8f c = {0.f, 0.f, 0.f, 0.f, 0.f, 0.f, 0.f, 0.f};
    const float* ar = &s_e[(i0 + rm) * Dn];
    const float* br = &s_e[(j0 + rm) * Dn];
#pragma unroll
    for (int s = 0; s < 6; s++) {           // K = 24 = 6 * 4
      v2f a, bb;
      a.x = ar[s * 4 + kb]; a.y = ar[s * 4 + kb + 1];
      bb.x = br[s * 4 + kb]; bb.y = br[s * 4 + kb + 1];
      c = __builtin_amdgcn_wmma_f32_16x16x4_f32(false, a, false, bb,
                                                (short)0, c, false, false);
    }
    const int ibase = i0 + ((lane >> 4) << 3);   // D layout: M = r + 8*(lane/16)
    const int j = j0 + rm;                       // N = lane % 16
    const float mj = s_m[j];
    const float sqj = s_sq[j];
#pragma unroll
    for (int r = 0; r < 8; r++) {
      int i = ibase + r;
      float pd2 = s_sq[i] + sqj - 2.f * c[r];
      pd2 = fminf(fmaxf(pd2, 0.f), 50.f);
      lsum += expf(-pd2) * s_m[i] * mj;
    }
  }
  lsum = wred(lsum);
  if ((tid & 31) == 0) s_red[wv] = lsum;
  __syncthreads();
  if (tid == 0) {
    float t = 0.f;
    for (int i = 0; i < 8; i++) t += s_red[i];
    wsf[O_SC + b * 16 + 8] = t;
  }
}

// ---------------- kernel 7: final reduction over batches ----------------
__global__ __launch_bounds__(32) void k_final(const float* wsf, float* out) {
  const int b = threadIdx.x;
  float tt = 0.f, bl = 0.f, al = 0.f, rl = 0.f, vv = 0.f;
  if (b < Bn) {
    const float* sc = wsf + O_SC + b * 16;
    float pm = sc[0], nm = sc[1], pc = sc[2], nc = sc[3];
    float ces = sc[4], slc = sc[5], attr = sc[6], mc = sc[7], rep = sc[8];
    float ce = ces / fmaxf(slc, 1.f);
    float bloss = ce + 5.f * (pm / fmaxf(pc, 1.f) + nm / fmaxf(nc, 1.f));
    float repl = (mc > 1.f) ? 1.5f * rep / fmaxf(mc * mc, 1.f) : 0.f;
    float total = bloss + attr + repl;
    bool valid = (pc >= 1.f) && (nc >= 1.f) && (slc > 0.f);
    float v = valid ? 1.f : 0.f;
    tt = total * v; bl = bloss * v; al = attr * v; rl = repl * v; vv = v;
  }
  tt = wred(tt); bl = wred(bl); al = wred(al); rl = wred(rl); vv = wred(vv);
  if (b == 0) {
    float safe = fmaxf(vv, 1.f);
    bool ok = vv > 0.f;
    out[0] = ok ? tt / safe : 0.f;
    out[1] = ok ? bl / safe : 0.f;
    out[2] = ok ? al / safe : 0.f;
    out[3] = ok ? rl / safe : 0.f;
  }
}

extern "C" void kernel_launch(void* const* d_in, const int* in_sizes, int n_in,
                              void* d_out, int out_size, void* d_ws, size_t ws_size,
                              hipStream_t stream) {
  const float* beta  = (const float*)d_in[0];
  const float* embed = (const float*)d_in[1];
  const int*   sid   = (const int*)d_in[2];
  const int*   cp    = (const int*)d_in[3];
  float*    out = (float*)d_out;
  float*    wsf = (float*)d_ws;
  unsigned* wsu = (unsigned*)d_ws;

  k_init<<<(O_END + 255) / 256, 256, 0, stream>>>(wsu);
  k_statsA<<<Bn * 64, 256, 0, stream>>>(beta, sid, cp, wsf, wsu);
  k_statsB<<<Bn * 64, 256, 0, stream>>>(beta, sid, wsf, wsu);
  k_gather<<<Bn, 512, 0, stream>>>(embed, wsf, wsu);
  k_attract<<<Bn * 64, 256, 0, stream>>>(embed, sid, wsf);
  k_finalize<<<Bn, 512, 0, stream>>>(wsf, wsu);
  k_repulse<<<Bn, 256, 0, stream>>>(wsf);
  k_final<<<1, 32, 0, stream>>>(wsf, out);
}